// MultiHeadedAttentionMix_45260365365793
// MI455X (gfx1250) — compile-verified
//
#include <hip/hip_runtime.h>
#include <hip/hip_bf16.h>

#define HEADS 8
#define DK 64
#define S 48
#define NQ (S*S)            // 2304
#define DMODEL (HEADS*DK)   // 512
#define BATCH 2
#define BH (BATCH*HEADS)    // 16

typedef __attribute__((ext_vector_type(2))) float v2f;
typedef __attribute__((ext_vector_type(8))) float v8f;

// ---------------------------------------------------------------------------
// Kernel 1: projection.  out[b,h,n,dk] = x[b,n,:] @ W^T + bias
// Register-blocked: one wave computes a 32(M=n) x 64(N=outfeat) strip using
// V_WMMA_F32_16X16X4_F32 (full fp32 matrix core). Per K-step: 2 A-frag loads
// + 4 B-frag loads feed 8 WMMAs (10.7 flop/byte from cache).
// The 64-wide N strip lies exactly inside one head -> trivial scatter.
// A lane layout (16x4 f32): lanes 0-15 hold K={0,1}, lanes 16-31 hold K={2,3}.
// ---------------------------------------------------------------------------
__global__ __launch_bounds__(256) void proj_kernel(const float* __restrict__ x,
                                                   const float* __restrict__ W,
                                                   const float* __restrict__ bias,
                                                   float* __restrict__ out)
{
    const int lane  = threadIdx.x & 31;
    const int wid   = blockIdx.x * 8 + (threadIdx.x >> 5);   // 1152 waves total
    const int nt    = wid & 7;             // N strip (512/64 = 8) == head index
    const int mt    = (wid >> 3) % 72;     // M strip (2304/32 = 72)
    const int b     = wid / (8 * 72);      // batch

    const int l15   = lane & 15;
    const int khalf = lane >> 4;           // 0: K={0,1}, 1: K={2,3}

    const float* xrow0 = x + ((size_t)b * NQ + mt * 32 + l15) * DMODEL;
    const float* xrow1 = xrow0 + 16 * DMODEL;
    const float* wrow0 = W + (size_t)(nt * 64 + l15) * DMODEL;
    const float* wrow1 = wrow0 + 16 * DMODEL;
    const float* wrow2 = wrow0 + 32 * DMODEL;
    const float* wrow3 = wrow0 + 48 * DMODEL;

    v8f c00 = {}, c01 = {}, c02 = {}, c03 = {};
    v8f c10 = {}, c11 = {}, c12 = {}, c13 = {};
    #pragma unroll 4
    for (int kb = 0; kb < DMODEL; kb += 4) {
        const int ko = kb + 2 * khalf;
        v2f a0 = *(const v2f*)(xrow0 + ko);
        v2f a1 = *(const v2f*)(xrow1 + ko);
        v2f b0 = *(const v2f*)(wrow0 + ko);
        v2f b1 = *(const v2f*)(wrow1 + ko);
        v2f b2 = *(const v2f*)(wrow2 + ko);
        v2f b3 = *(const v2f*)(wrow3 + ko);
        c00 = __builtin_amdgcn_wmma_f32_16x16x4_f32(false, a0, false, b0, (short)0, c00, false, false);
        c01 = __builtin_amdgcn_wmma_f32_16x16x4_f32(false, a0, false, b1, (short)0, c01, false, false);
        c02 = __builtin_amdgcn_wmma_f32_16x16x4_f32(false, a0, false, b2, (short)0, c02, false, false);
        c03 = __builtin_amdgcn_wmma_f32_16x16x4_f32(false, a0, false, b3, (short)0, c03, false, false);
        c10 = __builtin_amdgcn_wmma_f32_16x16x4_f32(false, a1, false, b0, (short)0, c10, false, false);
        c11 = __builtin_amdgcn_wmma_f32_16x16x4_f32(false, a1, false, b1, (short)0, c11, false, false);
        c12 = __builtin_amdgcn_wmma_f32_16x16x4_f32(false, a1, false, b2, (short)0, c12, false, false);
        c13 = __builtin_amdgcn_wmma_f32_16x16x4_f32(false, a1, false, b3, (short)0, c13, false, false);
    }

    // D layout: VGPR r, lane L -> M = r + 8*(L>>4), N = L&15.
    // Whole strip is head h = nt; dkk = j*16 + l15.
    v8f acc[2][4] = {{c00, c01, c02, c03}, {c10, c11, c12, c13}};
    float* obase = out + ((size_t)(b * HEADS + nt)) * NQ * DK;
    #pragma unroll
    for (int j = 0; j < 4; ++j) {
        int dkk  = j * 16 + l15;
        float bv = bias[nt * 64 + dkk];
        #pragma unroll
        for (int i = 0; i < 2; ++i) {
            #pragma unroll
            for (int r = 0; r < 8; ++r) {
                int grow = mt * 32 + i * 16 + r + 8 * khalf;
                obase[(size_t)grow * DK + dkk] = acc[i][j][r] + bv;
            }
        }
    }
}

// ---------------------------------------------------------------------------
// Kernel 2: scores[Bh][q][k] = sum_d q[Bh][q][d] * k[Bh][k][d]   (K = 64)
// Register-blocked 32(M) x 64(N) per wave: 128 fp32 WMMAs, 96 b64 loads.
// ---------------------------------------------------------------------------
__global__ __launch_bounds__(256) void scores_kernel(const float* __restrict__ q,
                                                     const float* __restrict__ k,
                                                     float* __restrict__ scores)
{
    const int lane  = threadIdx.x & 31;
    const int wid   = blockIdx.x * 8 + (threadIdx.x >> 5);   // 41472 waves
    const int nt    = wid % 36;              // key strip (2304/64 = 36)
    const int mt    = (wid / 36) % 72;       // query strip (2304/32 = 72)
    const int Bh    = wid / (36 * 72);

    const int l15   = lane & 15;
    const int khalf = lane >> 4;

    const float* qrow0 = q + ((size_t)Bh * NQ + mt * 32 + l15) * DK;
    const float* qrow1 = qrow0 + 16 * DK;
    const float* krow0 = k + ((size_t)Bh * NQ + nt * 64 + l15) * DK;
    const float* krow1 = krow0 + 16 * DK;
    const float* krow2 = krow0 + 32 * DK;
    const float* krow3 = krow0 + 48 * DK;

    v8f c00 = {}, c01 = {}, c02 = {}, c03 = {};
    v8f c10 = {}, c11 = {}, c12 = {}, c13 = {};
    #pragma unroll 4
    for (int kb = 0; kb < DK; kb += 4) {
        const int ko = kb + 2 * khalf;
        v2f a0 = *(const v2f*)(qrow0 + ko);
        v2f a1 = *(const v2f*)(qrow1 + ko);
        v2f b0 = *(const v2f*)(krow0 + ko);
        v2f b1 = *(const v2f*)(krow1 + ko);
        v2f b2 = *(const v2f*)(krow2 + ko);
        v2f b3 = *(const v2f*)(krow3 + ko);
        c00 = __builtin_amdgcn_wmma_f32_16x16x4_f32(false, a0, false, b0, (short)0, c00, false, false);
        c01 = __builtin_amdgcn_wmma_f32_16x16x4_f32(false, a0, false, b1, (short)0, c01, false, false);
        c02 = __builtin_amdgcn_wmma_f32_16x16x4_f32(false, a0, false, b2, (short)0, c02, false, false);
        c03 = __builtin_amdgcn_wmma_f32_16x16x4_f32(false, a0, false, b3, (short)0, c03, false, false);
        c10 = __builtin_amdgcn_wmma_f32_16x16x4_f32(false, a1, false, b0, (short)0, c10, false, false);
        c11 = __builtin_amdgcn_wmma_f32_16x16x4_f32(false, a1, false, b1, (short)0, c11, false, false);
        c12 = __builtin_amdgcn_wmma_f32_16x16x4_f32(false, a1, false, b2, (short)0, c12, false, false);
        c13 = __builtin_amdgcn_wmma_f32_16x16x4_f32(false, a1, false, b3, (short)0, c13, false, false);
    }

    v8f acc[2][4] = {{c00, c01, c02, c03}, {c10, c11, c12, c13}};
    const size_t base = (size_t)Bh * NQ * NQ;
    #pragma unroll
    for (int i = 0; i < 2; ++i) {
        #pragma unroll
        for (int r = 0; r < 8; ++r) {
            int grow = mt * 32 + i * 16 + r + 8 * khalf;
            float* srow = scores + base + (size_t)grow * NQ + nt * 64 + l15;
            #pragma unroll
            for (int j = 0; j < 4; ++j)
                srow[j * 16] = acc[i][j][r];
        }
    }
}

// ---------------------------------------------------------------------------
// Kernel 3: per-key-position argmax over the 2304 query positions.
// ---------------------------------------------------------------------------
__global__ void argmax_kernel(const float* __restrict__ scores, int* __restrict__ idx)
{
    int t = blockIdx.x * blockDim.x + threadIdx.x;
    if (t >= BH * NQ) return;
    int Bh = t / NQ, c = t % NQ;
    const float* col = scores + (size_t)Bh * NQ * NQ + c;
    float best = col[0];
    int bi = 0;
    for (int qq = 1; qq < NQ; ++qq) {
        float v = col[(size_t)qq * NQ];
        if (v > best) { best = v; bi = qq; }   // strict > keeps first max (jnp.argmax)
    }
    idx[t] = bi;
}

__device__ __forceinline__ float gauss_fac(float qx, float qy, int id)
{
    float ix = (float)(id % S), iy = (float)(id / S);
    float dx = qx - ix, dy = qy - iy;
    return __expf(-(dx * dx + dy * dy) * 0.02f);   // 1/(2*sigma^2) = 1/50
}

// ---------------------------------------------------------------------------
// Kernel 4: one wave32 per (Bh, q) row: max, then Z and x0 dot.
// ---------------------------------------------------------------------------
__global__ __launch_bounds__(256) void rowsoft_kernel(const float* __restrict__ scores,
                                                      const int* __restrict__ idx,
                                                      const float* __restrict__ value0,
                                                      float* __restrict__ rowm,
                                                      float* __restrict__ rowz,
                                                      float* __restrict__ x0)
{
    const int lane = threadIdx.x & 31;
    const int wid  = blockIdx.x * 8 + (threadIdx.x >> 5);    // 36864 rows
    const int Bh   = wid / NQ;
    const int qrow = wid % NQ;
    const int b    = Bh >> 3;

    const float qx = (2.0f / 47.0f) * (float)(qrow % S) - 1.0f;  // linspace(-1,1,48)
    const float qy = (2.0f / 47.0f) * (float)(qrow / S) - 1.0f;
    const float* srow = scores + ((size_t)Bh * NQ + qrow) * NQ;
    const int*   irow = idx + Bh * NQ;

    float m = -3.4e38f;
    for (int c = lane; c < NQ; c += 32) {
        float v = srow[c] * gauss_fac(qx, qy, irow[c]) * 0.125f;
        m = fmaxf(m, v);
    }
    for (int off = 16; off; off >>= 1) m = fmaxf(m, __shfl_xor(m, off, 32));

    float z = 0.0f, dot = 0.0f;
    const float* v0 = value0 + b * NQ;
    for (int c = lane; c < NQ; c += 32) {
        float v = srow[c] * gauss_fac(qx, qy, irow[c]) * 0.125f;
        float e = __expf(v - m);
        z   += e;
        dot += e * v0[c];
    }
    for (int off = 16; off; off >>= 1) {
        z   += __shfl_xor(z, off, 32);
        dot += __shfl_xor(dot, off, 32);
    }
    if (lane == 0) {
        rowm[wid] = m;
        rowz[wid] = z;
        x0[wid]   = dot / z;
    }
}

// ---------------------------------------------------------------------------
// Kernel 5: x1[Bh][c] = sum_q attn[q][c] * value1[b][q]  (column reduction).
// ---------------------------------------------------------------------------
__global__ void x1_kernel(const float* __restrict__ scores, const int* __restrict__ idx,
                          const float* __restrict__ rowm, const float* __restrict__ rowz,
                          const float* __restrict__ value1, float* __restrict__ x1)
{
    int t = blockIdx.x * blockDim.x + threadIdx.x;
    if (t >= BH * NQ) return;
    int Bh = t / NQ, c = t % NQ;
    int b = Bh >> 3;
    int id = idx[t];
    float ix = (float)(id % S), iy = (float)(id / S);
    const float* col = scores + (size_t)Bh * NQ * NQ + c;
    const float* m   = rowm + Bh * NQ;
    const float* z   = rowz + Bh * NQ;
    const float* v1  = value1 + b * NQ;

    float acc = 0.0f;
    for (int qq = 0; qq < NQ; ++qq) {
        float qx = (2.0f / 47.0f) * (float)(qq % S) - 1.0f;
        float qy = (2.0f / 47.0f) * (float)(qq / S) - 1.0f;
        float dx = qx - ix, dy = qy - iy;
        float gk = __expf(-(dx * dx + dy * dy) * 0.02f);
        float e  = __expf(col[(size_t)qq * NQ] * gk * 0.125f - m[qq]) / z[qq];
        acc += e * v1[qq];
    }
    x1[t] = acc;
}

// ---------------------------------------------------------------------------
// Kernel 6: out[b][q][c] = (1/HEADS) * sum_h x0[b,h,q] * x1[b,h,c]
// ---------------------------------------------------------------------------
__global__ void outer_kernel(const float* __restrict__ x0, const float* __restrict__ x1,
                             float* __restrict__ out)
{
    int t = blockIdx.x * blockDim.x + threadIdx.x;   // up to 10,616,832 < 2^31
    if (t >= BATCH * NQ * NQ) return;
    int c    = t % NQ;
    int qrow = (t / NQ) % NQ;
    int b    = t / (NQ * NQ);
    float acc = 0.0f;
    #pragma unroll
    for (int h = 0; h < HEADS; ++h)
        acc += x0[(b * HEADS + h) * NQ + qrow] * x1[(b * HEADS + h) * NQ + c];
    out[t] = acc * (1.0f / HEADS);
}

// ---------------------------------------------------------------------------
extern "C" void kernel_launch(void* const* d_in, const int* in_sizes, int n_in,
                              void* d_out, int out_size, void* d_ws, size_t ws_size,
                              hipStream_t stream)
{
    const float* query  = (const float*)d_in[0];
    const float* key    = (const float*)d_in[1];
    const float* value0 = (const float*)d_in[2];
    const float* value1 = (const float*)d_in[3];
    const float* Wq     = (const float*)d_in[4];
    const float* bq     = (const float*)d_in[5];
    const float* Wk     = (const float*)d_in[6];
    const float* bk     = (const float*)d_in[7];
    float* out = (float*)d_out;

    // Workspace layout (floats): q | k | scores | idx | rowm | rowz | x0 | x1
    float* ws = (float*)d_ws;
    size_t off = 0;
    float* qbuf   = ws + off; off += (size_t)BATCH * HEADS * NQ * DK;  // 9.4 MB
    float* kbuf   = ws + off; off += (size_t)BATCH * HEADS * NQ * DK;  // 9.4 MB
    float* scores = ws + off; off += (size_t)BH * NQ * NQ;             // 340 MB
    int*   idxb   = (int*)(ws + off); off += (size_t)BH * NQ;
    float* rowm   = ws + off; off += (size_t)BH * NQ;
    float* rowz   = ws + off; off += (size_t)BH * NQ;
    float* x0     = ws + off; off += (size_t)BH * NQ;
    float* x1     = ws + off; off += (size_t)BH * NQ;

    // 1+2) projections: 2 * 72 * 8 = 1152 waves, 8 waves/block
    proj_kernel<<<144, 256, 0, stream>>>(query, Wq, bq, qbuf);
    proj_kernel<<<144, 256, 0, stream>>>(key,   Wk, bk, kbuf);
    // 3) scores: 16 * 72 * 36 = 41472 waves, 8 waves/block
    scores_kernel<<<5184, 256, 0, stream>>>(qbuf, kbuf, scores);
    // 4) column argmax
    argmax_kernel<<<(BH * NQ + 255) / 256, 256, 0, stream>>>(scores, idxb);
    // 5) row softmax + x0 (one wave per row; 36864 rows / 8 waves)
    rowsoft_kernel<<<(BH * NQ) / 8, 256, 0, stream>>>(scores, idxb, value0,
                                                      rowm, rowz, x0);
    // 6) x1 column pass
    x1_kernel<<<(BH * NQ + 255) / 256, 256, 0, stream>>>(scores, idxb, rowm, rowz,
                                                         value1, x1);
    // 7) head-averaged outer product
    int total = BATCH * NQ * NQ;
    outer_kernel<<<(total + 255) / 256, 256, 0, stream>>>(x0, x1, out);
}